// LocallyConnected2D_42348377539022
// MI455X (gfx1250) — compile-verified
//
#include <hip/hip_runtime.h>

typedef float v2f __attribute__((ext_vector_type(2)));
typedef float v8f __attribute__((ext_vector_type(8)));
typedef unsigned int v4u __attribute__((ext_vector_type(4)));
typedef int v4i __attribute__((ext_vector_type(4)));
typedef int v8i __attribute__((ext_vector_type(8)));

#define B_    16
#define IH_   32
#define IW_   32
#define CIN_  64
#define COUT_ 64
#define OH_   32
#define OW_   32
#define KHW_  9

#define JT     4                   // j positions per block
#define CC     4                   // input channels per K-chunk
#define KCHUNK (CC * KHW_)         // 36 K values per chunk
#define NCHUNK (CIN_ / CC)         // 16 chunks
#define MROW   (JT * KHW_)         // 36: contiguous (j',k') run per (o,c')

#define AROW 38                    // padded LDS row for patches (even => 8B-aligned pairs)

#define WCHUNK_FLOATS (CC * COUT_ * MROW)      // 9216
#define WCHUNK_BYTES  (WCHUNK_FLOATS * 4)      // 36864
#define ACHUNK_FLOATS (JT * B_ * AROW)         // 2432
#define LDS_BYTES     (2 * WCHUNK_BYTES + ACHUNK_FLOATS * 4)   // 83456

#define OSTRIDE (CIN_ * OH_ * OW_ * KHW_)      // 589824 (o stride, elements)
#define CSTRIDE (OH_ * OW_ * KHW_)             // 9216   (c stride, elements)

// ---- TDM availability / arity selection (device pass only) ----
#if defined(__AMDGCN__) && __has_include(<hip/amd_detail/amd_gfx1250_TDM.h>)
  #define TDM_ARITY 6
#elif defined(__AMDGCN__) && __has_builtin(__builtin_amdgcn_tensor_load_to_lds)
  #define TDM_ARITY 5
#else
  #define TDM_ARITY 0
#endif

#if TDM_ARITY > 0
// One TDM descriptor = the whole weight chunk for this block:
//   dim0: 36 contiguous floats (j'*9+k'),  dim1: o (stride 589824), dim2: c' (stride 9216)
// LDS fill order: ((c' * 64 + o) * 36 + m), contiguous from lds_off.
__device__ __forceinline__ void tdm_load_wchunk(const float* w, int i, int j0,
                                                int cbase, unsigned lds_off) {
  unsigned long long ga = (unsigned long long)(uintptr_t)w +
      4ull * ((unsigned long long)cbase * CSTRIDE + (unsigned long long)i * (OW_ * KHW_) +
              (unsigned long long)j0 * KHW_);
  v4u g0;
  g0.x = 1u;                                               // count=1, user mode
  g0.y = lds_off;                                          // lds_addr (bytes)
  g0.z = (unsigned)ga;                                     // global_addr[31:0]
  g0.w = (unsigned)((ga >> 32) & 0x01FFFFFFull) | (2u << 30);  // addr[56:32] | type=2

  v8i g1 = {
    (int)(2u << 16),                 // w0: wg_mask=0 | data_size=2 (4B)
    (int)(36u << 16),                // w1: tensor_dim0[15:0]=36
    (int)(64u << 16),                // w2: tensor_dim1[15:0]=64 (o)
    (int)(36u << 16),                // w3: tile_dim0=36
    (int)(64u | (CC << 16)),         // w4: tile_dim1=64 (o) | tile_dim2=CC (c')
    (int)OSTRIDE,                    // w5: tensor_dim0_stride[31:0]=589824
    (int)((CSTRIDE & 0xFFFF) << 16), // w6: stride0[47:32]=0 | stride1[15:0]
    (int)(CSTRIDE >> 16)             // w7: tensor_dim1_stride[47:16]
  };
  v4i g2 = { CC, 0, 0, 0 };          // tensor_dim2=CC (c'); rest unused
  v4i g3 = { 0, 0, 0, 0 };
#if TDM_ARITY == 5
  __builtin_amdgcn_tensor_load_to_lds(g0, g1, g2, g3, 0);
#else
  v8i g4 = { 0, 0, 0, 0, 0, 0, 0, 0 };
  __builtin_amdgcn_tensor_load_to_lds(g0, g1, g2, g3, g4, 0);
#endif
}
#endif

__global__ __launch_bounds__(256) void lc2d_wmma_tdm_kernel(
    const float* __restrict__ x, const float* __restrict__ w,
    const float* __restrict__ bias, float* __restrict__ out)
{
  extern __shared__ float smem[];
  float* wbuf0 = smem;                          // [c'][o][36]
  float* wbuf1 = smem + WCHUNK_FLOATS;
  float* sA    = smem + 2 * WCHUNK_FLOATS;      // [(j'*16+b)][AROW]

  const int tid  = threadIdx.x;
  const int lane = tid & 31;
  const int wid  = tid >> 5;          // 0..7
  const int nt   = wid & 3;           // COUT tile (o / 16)
  const int jp   = wid >> 2;          // 0..1 -> j' pair {2jp, 2jp+1}
  const int bx   = blockIdx.x;        // 0..255
  const int i    = bx >> 3;
  const int j0   = (bx & 7) * JT;

  const int halfK = (lane >> 4) << 1;
  const int mrow  = lane & 15;
  const int oidx  = nt * 16 + mrow;

  v8f acc0 = {0.f, 0.f, 0.f, 0.f, 0.f, 0.f, 0.f, 0.f};
  v8f acc1 = acc0;

#if TDM_ARITY > 0
  const unsigned ldsbase = (unsigned)(unsigned long long)(uintptr_t)(void*)smem;
  if (wid == 0)
    tdm_load_wchunk(w, i, j0, 0, ldsbase);          // prefetch chunk 0 -> wbuf0
#endif

  for (int ch = 0; ch < NCHUNK; ++ch) {
    const int cbase = ch * CC;
    float* sW = (ch & 1) ? wbuf1 : wbuf0;

    // ---- stage x patches for this chunk: 4j' x 16b x 36K (c' innermost) ----
    for (int f = tid; f < JT * B_ * KCHUNK; f += 256) {
      int c  = f & 3;
      int t  = f >> 2;
      int kq = t % KHW_;
      t /= KHW_;
      int b  = t & 15;
      int jq = t >> 4;
      int di = kq / 3, dj = kq - di * 3;
      int ih = i + di - 1;
      int iw = j0 + jq + dj - 1;
      float v = 0.f;
      if (ih >= 0 && ih < IH_ && iw >= 0 && iw < IW_)
        v = x[((b * IH_ + ih) * IW_ + iw) * CIN_ + cbase + c];
      sA[(jq * B_ + b) * AROW + c * KHW_ + kq] = v;
    }

#if TDM_ARITY > 0
    if (wid == 0) {
      if (ch + 1 < NCHUNK) {
        // async: fetch next chunk into the other buffer while we compute
        tdm_load_wchunk(w, i, j0, cbase + CC,
                        ldsbase + (unsigned)(((ch + 1) & 1) * WCHUNK_BYTES));
        __builtin_amdgcn_s_wait_tensorcnt(1);   // current chunk's TDM complete
      } else {
        __builtin_amdgcn_s_wait_tensorcnt(0);
      }
    }
#else
    // fallback: synchronous cooperative staging (144B coalesced runs)
    for (int f = tid; f < COUT_ * CC * MROW; f += 256) {
      int o  = f / (CC * MROW);
      int r  = f - o * (CC * MROW);
      int cp = r / MROW;
      int m  = r - cp * MROW;
      int jq = m / KHW_;
      int kq = m - jq * KHW_;
      sW[(cp * COUT_ + o) * MROW + m] =
          w[(size_t)o * OSTRIDE + (size_t)(cbase + cp) * CSTRIDE +
            i * (OW_ * KHW_) + (j0 + jq) * KHW_ + kq];
    }
#endif
    __syncthreads();   // publish sA (+ TDM-filled sW) to all waves

    // ---- 9 k-steps of V_WMMA_F32_16X16X4_F32 per j' (2 independent chains) ----
#pragma unroll
    for (int kk = 0; kk < KCHUNK; kk += 4) {
      const int K0 = kk + halfK;
      v2f aA0 = *(const v2f*)&sA[((2 * jp + 0) * B_ + mrow) * AROW + K0];
      v2f aA1 = *(const v2f*)&sA[((2 * jp + 1) * B_ + mrow) * AROW + K0];
      int c0 = K0 / 9,       k0 = K0 - c0 * 9;
      int c1 = (K0 + 1) / 9, k1 = (K0 + 1) - c1 * 9;
      v2f bB0, bB1;
      bB0.x = sW[(c0 * COUT_ + oidx) * MROW + (2 * jp + 0) * KHW_ + k0];
      bB0.y = sW[(c1 * COUT_ + oidx) * MROW + (2 * jp + 0) * KHW_ + k1];
      bB1.x = sW[(c0 * COUT_ + oidx) * MROW + (2 * jp + 1) * KHW_ + k0];
      bB1.y = sW[(c1 * COUT_ + oidx) * MROW + (2 * jp + 1) * KHW_ + k1];
      acc0 = __builtin_amdgcn_wmma_f32_16x16x4_f32(
          false, aA0, false, bB0, (short)0, acc0, false, false);
      acc1 = __builtin_amdgcn_wmma_f32_16x16x4_f32(
          false, aA1, false, bB1, (short)0, acc1, false, false);
    }
    __syncthreads();   // all reads of sA / sW done before next-iter overwrite
  }

  // ---- epilogue: D layout VGPR v -> (M = v + 8*(lane>=16), N = lane&15) ----
  const int bhi = (lane >> 4) * 8;
#pragma unroll
  for (int jj = 0; jj < 2; ++jj) {
    const int j = j0 + 2 * jp + jj;
    const v8f& a = jj ? acc1 : acc0;
    const float bv = bias[(i * OW_ + j) * COUT_ + oidx];
#pragma unroll
    for (int v = 0; v < 8; ++v) {
      const int brow = v + bhi;
      out[((brow * OH_ + i) * OW_ + j) * COUT_ + oidx] = a[v] + bv;
    }
  }
}

extern "C" void kernel_launch(void* const* d_in, const int* in_sizes, int n_in,
                              void* d_out, int out_size, void* d_ws, size_t ws_size,
                              hipStream_t stream) {
  const float* x    = (const float*)d_in[0];
  const float* w    = (const float*)d_in[1];
  const float* bias = (const float*)d_in[2];
  float* out        = (float*)d_out;
  dim3 grid(256);     // 32 rows x 8 j-tiles
  dim3 block(256);    // 8 wave32
  hipLaunchKernelGGL(lc2d_wmma_tdm_kernel, grid, block, LDS_BYTES, stream,
                     x, w, bias, out);
}